// ConsPosiEmb_40192303956653
// MI455X (gfx1250) — compile-verified
//
#include <hip/hip_runtime.h>
#include <hip/hip_bf16.h>
#include <stdint.h>
#include <stddef.h>

// ---------------------------------------------------------------------------
// ConsPosiEmb for MI455X (gfx1250): position scan + table gather.
//   tokens : [32, 4096] int32   (d_in[0])
//   weights: [4098, 1024] fp32  (d_in[1])
//   out    : [32, 4096, 1024] fp32
// Store-bandwidth bound: ~512 MiB out @ 23.3 TB/s ~= 23 us floor.
// Weights (16 MiB) are kept L2-resident; output uses non-temporal stores.
// ---------------------------------------------------------------------------

#define PAD_IDX 1
#define SEP_ID  4
#define BSZ     32
#define SEQ     4096
#define DIM     1024
#define NROWS   (PAD_IDX + 1 + SEQ)   // 4098 table rows

typedef __attribute__((ext_vector_type(4))) float v4f;
typedef __attribute__((ext_vector_type(4))) int   v4i;
// GCC-style vector: matches the async-LDS builtin's parameter type (V4i).
typedef int g4i __attribute__((vector_size(16)));

#if defined(__AMDGCN__) &&                                                    \
    __has_builtin(__builtin_amdgcn_global_load_async_to_lds_b128) &&          \
    __has_builtin(__builtin_amdgcn_s_wait_asynccnt)
#define USE_ASYNC_LDS 1
#else
#define USE_ASYNC_LDS 0
#endif

// ---------------------------------------------------------------------------
// Kernel A: one block per batch row. Computes
//   first_pad = min{ j : tok[j]==PAD }            (block min-reduction)
//   last_sep[j] = inclusive prefix-max of (tok==SEP ? j : 0)
//   pos[j] = (j < first_pad) ? j - last_sep[j] + 2 : 1
// Token tile is staged global->LDS via the gfx1250 async path (ASYNCcnt);
// while it is in flight, the grid prefetches the 16 MiB table into L2 once.
// ---------------------------------------------------------------------------
extern "C" __global__ __launch_bounds__(1024)
void cpe_positions_kernel(const int* __restrict__ tokens,
                          const float* __restrict__ wtab,
                          int* __restrict__ pos)
{
    __shared__ v4i stok4[SEQ / 4];   // 16 KB: this row's tokens
    __shared__ int ssep[1024];       //  4 KB: per-thread last-sep scan
    __shared__ int s_firstpad;

    const int    t    = threadIdx.x;
    const int    j0   = t * 4;
    const size_t base = (size_t)blockIdx.x * SEQ;

#if USE_ASYNC_LDS
    // global_load_async_to_lds_b128: 16 B per lane, tracked on ASYNCcnt.
    __builtin_amdgcn_global_load_async_to_lds_b128(
        (__attribute__((address_space(1))) g4i*)(tokens + base + j0),
        (__attribute__((address_space(3))) g4i*)&stok4[t],
        0, 0);
#else
    stok4[t] = *(const v4i*)(tokens + base + j0);
#endif

    // Warm L2 with the embedding table while the token tile is in flight:
    // 32768 threads x 512 B = one distributed pass over the first 16 MiB of
    // the ~16.01 MiB table (max offset 16,777,088 < 16,785,408, so no bounds
    // check is needed -> branchless, 4 unconditional global_prefetch_b8).
    {
        const char*  wb  = (const char*)wtab;
        const size_t off = ((size_t)blockIdx.x * 1024u + (size_t)t) * 512u;
#pragma unroll
        for (int k = 0; k < 4; ++k) {
            __builtin_prefetch(wb + off + (size_t)k * 128u, 0, 1);
        }
    }

    if (t == 0) s_firstpad = 0x7fffffff;
#if USE_ASYNC_LDS
    __builtin_amdgcn_s_wait_asynccnt(0);   // own tile resident in LDS
#endif
    __syncthreads();

    const v4i tk = stok4[t];

    // Per-thread summaries over 4 consecutive tokens.
    int firstpad = 0x7fffffff;
    if (tk.w == PAD_IDX) firstpad = j0 + 3;
    if (tk.z == PAD_IDX) firstpad = j0 + 2;
    if (tk.y == PAD_IDX) firstpad = j0 + 1;
    if (tk.x == PAD_IDX) firstpad = j0 + 0;

    int maxsep = 0;
    if (tk.x == SEP_ID) maxsep = j0 + 0;
    if (tk.y == SEP_ID) maxsep = j0 + 1;
    if (tk.z == SEP_ID) maxsep = j0 + 2;
    if (tk.w == SEP_ID) maxsep = j0 + 3;

    atomicMin(&s_firstpad, firstpad);    // ds_min on LDS
    ssep[t] = maxsep;
    __syncthreads();

    // Hillis-Steele inclusive max-scan across the 1024 thread summaries.
    for (int off = 1; off < 1024; off <<= 1) {
        const int a = ssep[t];
        const int b = (t >= off) ? ssep[t - off] : 0;
        __syncthreads();
        ssep[t] = (b > a) ? b : a;
        __syncthreads();
    }

    const int first_pad = s_firstpad;
    int running = (t > 0) ? ssep[t - 1] : 0;   // exclusive prefix for this thread

    v4i r4;
    int j = j0;
    if (tk.x == SEP_ID) running = j;           // j exceeds every earlier index
    r4.x = (j < first_pad) ? (j - running + 2) : 1;
    j = j0 + 1;
    if (tk.y == SEP_ID) running = j;
    r4.y = (j < first_pad) ? (j - running + 2) : 1;
    j = j0 + 2;
    if (tk.z == SEP_ID) running = j;
    r4.z = (j < first_pad) ? (j - running + 2) : 1;
    j = j0 + 3;
    if (tk.w == SEP_ID) running = j;
    r4.w = (j < first_pad) ? (j - running + 2) : 1;

    *(v4i*)(pos + base + j0) = r4;             // global_store_b128
}

// ---------------------------------------------------------------------------
// Kernel B: one 4 KiB output row per 256-thread block.
//   - pos[r] is block-uniform -> readfirstlane pins it (and the src base) to
//     SGPRs; the row load is a single b128 per lane (L2 hit: table is resident).
//   - output is written with non-temporal b128 stores so the 512 MiB stream
//     does not evict the weights table from L2.
// ---------------------------------------------------------------------------
extern "C" __global__ __launch_bounds__(256)
void cpe_gather_kernel(const float* __restrict__ wtab,
                       const int* __restrict__ pos,
                       float* __restrict__ out)
{
    const int r = blockIdx.x;
    const int p = __builtin_amdgcn_readfirstlane(pos[r]);

    const v4f* __restrict__ src = (const v4f*)(wtab + (size_t)p * DIM);
    v4f*                    dst = (v4f*)(out + (size_t)r * DIM);

    const v4f v = src[threadIdx.x];                       // global_load_b128
    __builtin_nontemporal_store(v, dst + threadIdx.x);    // global_store_b128 (NT)
}

// ---------------------------------------------------------------------------
extern "C" void kernel_launch(void* const* d_in, const int* in_sizes, int n_in,
                              void* d_out, int out_size, void* d_ws, size_t ws_size,
                              hipStream_t stream)
{
    (void)in_sizes; (void)n_in; (void)out_size; (void)ws_size;

    const int*   tokens  = (const int*)d_in[0];
    const float* weights = (const float*)d_in[1];
    float*       out     = (float*)d_out;
    int*         pos     = (int*)d_ws;       // BSZ*SEQ int32 = 512 KiB scratch

    cpe_positions_kernel<<<dim3(BSZ), dim3(1024), 0, stream>>>(tokens, weights, pos);
    cpe_gather_kernel<<<dim3(BSZ * SEQ), dim3(256), 0, stream>>>(weights, pos, out);
}